// MultiTaskLoss_5703716569769
// MI455X (gfx1250) — compile-verified
//
#include <hip/hip_runtime.h>
#include <hip/hip_bf16.h>
#include <math.h>

#define Nn 8
#define Hh 512
#define Ww 512
#define PIX (Nn*Hh*Ww)   // 2097152
#define IMG (Hh*Ww)      // 262144
#define INF_D 1.0e6f
#define BIGMIN 3.0e12f

typedef float v2f __attribute__((ext_vector_type(2)));
typedef float v8f __attribute__((ext_vector_type(8)));

// ---------------- wave32 sum via V_WMMA_F32_16X16X4_F32 ----------------
// A (16x4 fp32) holds 64 partials: lane l<16 -> A[l,0]=x, A[l,1]=y;
// lane l>=16 -> A[l-16,2]=x, A[l-16,3]=y. B = ones(4x16) => D[m,n]=rowsum(A)[m].
// D layout: vgpr v, lane<16 -> D[v,lane]; lane>=16 -> D[v+8,lane-16].
// Sum of 8 D vgprs gives rows 0-7 (lanes 0-15) / rows 8-15 (lanes 16-31);
// one xor-16 shuffle completes the 32-lane total (uniform across lanes).
__device__ __forceinline__ float wave_sum32(float x, float y) {
#if defined(__has_builtin) && __has_builtin(__builtin_amdgcn_wmma_f32_16x16x4_f32)
    v2f a; a[0] = x; a[1] = y;
    v2f b; b[0] = 1.0f; b[1] = 1.0f;
    v8f c = {0.f,0.f,0.f,0.f,0.f,0.f,0.f,0.f};
    c = __builtin_amdgcn_wmma_f32_16x16x4_f32(false, a, false, b,
                                              (short)0, c, false, false);
    float t = c[0]+c[1]+c[2]+c[3]+c[4]+c[5]+c[6]+c[7];
    t += __shfl_xor(t, 16, 32);
    return t;
#else
    float t = x + y;
    for (int off = 16; off > 0; off >>= 1) t += __shfl_xor(t, off, 32);
    return t;
#endif
}

// ---------------- kernel 0: zero accumulators ----------------
__global__ void k_init(unsigned* maxs, float* acc) {
    int t = threadIdx.x;
    if (t < Nn) maxs[t] = 0u;          // encoded-float max, 0 == -inf-ish
    if (t < 27) acc[t] = 0.0f;         // 3 scalars + 3*8 per-image sums
}

// ---------------- kernel 1: 1D EDT along W (both polarities) ----------------
// g1[w] = min_j(g[j] + |w-j|) = min(w + prefmin(g[j]-j), -w + sufmin(g[j]+j))
__device__ __forceinline__ float pref_min(float v, float* lds, int w) {
    lds[w] = v; __syncthreads();
    for (int off = 1; off < Ww; off <<= 1) {
        float o = (w >= off) ? lds[w - off] : BIGMIN;
        __syncthreads();
        v = fminf(v, o);
        lds[w] = v;
        __syncthreads();
    }
    return v;
}
__device__ __forceinline__ float suf_min(float v, float* lds, int w) {
    lds[w] = v; __syncthreads();
    for (int off = 1; off < Ww; off <<= 1) {
        float o = (w + off < Ww) ? lds[w + off] : BIGMIN;
        __syncthreads();
        v = fminf(v, o);
        lds[w] = v;
        __syncthreads();
    }
    return v;
}

__global__ void k_row_edt(const float* __restrict__ gt,
                          float* __restrict__ g2_in,
                          float* __restrict__ g2_out) {
    __shared__ float lds[Ww];
    int row = blockIdx.x;                 // n*H + h
    int w   = threadIdx.x;
    int idx = row * Ww + w;
    bool m  = gt[idx] > 0.5f;
    float gin  = m ? INF_D : 0.0f;        // dist_in : feature = ~m
    float gout = m ? 0.0f : INF_D;        // dist_out: feature =  m
    float fw = (float)w;

    float p  = pref_min(gin  - fw, lds, w);
    float s  = suf_min (gin  + fw, lds, w);
    float g1i = fminf(fw + p, s - fw);
    p = pref_min(gout - fw, lds, w);
    s = suf_min (gout + fw, lds, w);
    float g1o = fminf(fw + p, s - fw);

    g2_in[idx]  = g1i * g1i;
    g2_out[idx] = g1o * g1o;
}

// ---------------- kernel 2: min-plus along H + sd + per-image max ----------------
#define KT 64
__global__ void k_minplus(const float* __restrict__ g2_in,
                          const float* __restrict__ g2_out,
                          float* __restrict__ sd,
                          unsigned* __restrict__ maxs) {
    __shared__ float lin [KT][32];
    __shared__ float lout[KT][32];
    __shared__ float red[256];

    int bid = blockIdx.x;
    int n  = bid >> 10;                 // / (16 w-tiles * 64 i-tiles)
    int r  = bid & 1023;
    int wt = r & 15;
    int it = r >> 4;
    int wl = threadIdx.x & 31;
    int il = threadIdx.x >> 5;          // 0..7
    int w  = wt * 32 + wl;
    int i  = it * 8  + il;

    float min_in = BIGMIN, min_out = BIGMIN;
    for (int k0 = 0; k0 < Hh; k0 += KT) {
        #pragma unroll
        for (int j = 0; j < 8; ++j) {
            int kk = il * 8 + j;
            int gidx = (n * Hh + (k0 + kk)) * Ww + w;   // coalesced over wl
            lin [kk][wl] = g2_in [gidx];
            lout[kk][wl] = g2_out[gidx];
        }
        __syncthreads();
        float dk = (float)(i - k0);
        #pragma unroll 8
        for (int kk = 0; kk < KT; ++kk) {
            float q = dk * dk;
            min_in  = fminf(min_in,  lin [kk][wl] + q);
            min_out = fminf(min_out, lout[kk][wl] + q);
            dk -= 1.0f;
        }
        __syncthreads();
    }
    float sdv = sqrtf(min_in) - sqrtf(min_out);
    sd[(n * Hh + i) * Ww + w] = sdv;

    // block max -> encoded atomic max per image
    red[threadIdx.x] = sdv; __syncthreads();
    for (int off = 128; off > 0; off >>= 1) {
        if (threadIdx.x < off)
            red[threadIdx.x] = fmaxf(red[threadIdx.x], red[threadIdx.x + off]);
        __syncthreads();
    }
    if (threadIdx.x == 0) {
        unsigned u = __float_as_uint(red[0]);
        u = (u & 0x80000000u) ? ~u : (u | 0x80000000u);
        atomicMax(&maxs[n], u);
    }
}

// ---------------- kernel 3: fused loss sums (WMMA wave reductions) ----------------
__global__ void k_loss(const float* __restrict__ seg_pred,
                       const float* __restrict__ seg_gt,
                       const float* __restrict__ edge_pred,
                       const float* __restrict__ edge_gt,
                       const float* __restrict__ sd,
                       const unsigned* __restrict__ maxs,
                       float* __restrict__ acc) {
    int n     = blockIdx.x >> 8;        // 256 blocks per image
    int chunk = blockIdx.x & 255;
    int base  = n * IMG + chunk * 1024 + threadIdx.x * 4;

    unsigned u = maxs[n];
    float mx  = (u & 0x80000000u) ? __uint_as_float(u ^ 0x80000000u)
                                  : __uint_as_float(~u);
    float inv = 1.0f / (mx + 1e-6f);
    bool norm = mx > 0.0f;

    float2 a_seg={0,0}, a_edg={0,0}, a_bnd={0,0},
           a_int={0,0}, a_sp={0,0},  a_sg={0,0};
    const float eps = 1e-7f;
    #pragma unroll
    for (int j = 0; j < 4; ++j) {
        int idx = base + j;
        float sp = seg_pred[idx],  sg = seg_gt[idx];
        float ep = edge_pred[idx], eg = edge_gt[idx];
        float s  = sd[idx];
        float pc = fminf(fmaxf(sp, eps), 1.0f - eps);
        float bs = -(sg * logf(pc) + (1.0f - sg) * log1pf(-pc));
        float ec = fminf(fmaxf(ep, eps), 1.0f - eps);
        float be = -(eg * logf(ec) + (1.0f - eg) * log1pf(-ec));
        be *= (eg > 0.5f) ? 0.9f : 0.1f;
        float sn = norm ? s * inv : s;
        sn = fminf(fmaxf(sn, -1.0f), 1.0f);
        float bd = fabsf(sp - sg) * fabsf(sn);
        float it = sp * sg;
        if (j & 1) { a_seg.y+=bs; a_edg.y+=be; a_bnd.y+=bd;
                     a_int.y+=it; a_sp.y+=sp;  a_sg.y+=sg; }
        else       { a_seg.x+=bs; a_edg.x+=be; a_bnd.x+=bd;
                     a_int.x+=it; a_sp.x+=sp;  a_sg.x+=sg; }
    }
    // wave32 totals via v_wmma_f32_16x16x4_f32 (uniform flow, EXEC all 1s)
    float t0 = wave_sum32(a_seg.x, a_seg.y);
    float t1 = wave_sum32(a_edg.x, a_edg.y);
    float t2 = wave_sum32(a_bnd.x, a_bnd.y);
    float t3 = wave_sum32(a_int.x, a_int.y);
    float t4 = wave_sum32(a_sp.x,  a_sp.y);
    float t5 = wave_sum32(a_sg.x,  a_sg.y);

    __shared__ float wsum[8][6];
    int lane = threadIdx.x & 31, wv = threadIdx.x >> 5;
    if (lane == 0) {
        wsum[wv][0]=t0; wsum[wv][1]=t1; wsum[wv][2]=t2;
        wsum[wv][3]=t3; wsum[wv][4]=t4; wsum[wv][5]=t5;
    }
    __syncthreads();
    if (threadIdx.x == 0) {
        float s0=0,s1=0,s2=0,s3=0,s4=0,s5=0;
        for (int k = 0; k < 8; ++k) {
            s0+=wsum[k][0]; s1+=wsum[k][1]; s2+=wsum[k][2];
            s3+=wsum[k][3]; s4+=wsum[k][4]; s5+=wsum[k][5];
        }
        atomicAdd(&acc[0], s0);        // seg bce sum
        atomicAdd(&acc[1], s1);        // edge weighted bce sum
        atomicAdd(&acc[2], s2);        // boundary sum
        atomicAdd(&acc[3  + n], s3);   // inter[n]
        atomicAdd(&acc[11 + n], s4);   // sum_pred[n]
        atomicAdd(&acc[19 + n], s5);   // sum_gt[n]
    }
}

// ---------------- kernel 4: finalize ----------------
__global__ void k_final(const float* __restrict__ acc, float* __restrict__ out) {
    if (threadIdx.x == 0 && blockIdx.x == 0) {
        const float invPix = 1.0f / (float)PIX;
        float seg_bce = acc[0] * invPix;
        float dsum = 0.0f;
        for (int n = 0; n < Nn; ++n) {
            float inter = acc[3 + n], p = acc[11 + n], g = acc[19 + n];
            dsum += 2.0f * inter / (p + g + 1e-6f);
        }
        float dice = 1.0f - dsum / (float)Nn;
        float edge = acc[1] * invPix * 0.5f;
        float bnd  = 0.2f * (acc[2] * invPix);
        out[0] = seg_bce + 0.5f * dice + edge + bnd;
    }
}

extern "C" void kernel_launch(void* const* d_in, const int* in_sizes, int n_in,
                              void* d_out, int out_size, void* d_ws, size_t ws_size,
                              hipStream_t stream) {
    const float* seg_pred  = (const float*)d_in[0];
    const float* seg_gt    = (const float*)d_in[1];
    const float* edge_pred = (const float*)d_in[2];
    const float* edge_gt   = (const float*)d_in[3];
    float* out = (float*)d_out;

    // workspace layout: g2_in | g2_out | sd | 8 uint maxs | 27 float acc
    float* g2_in  = (float*)d_ws;
    float* g2_out = g2_in  + PIX;
    float* sd     = g2_out + PIX;
    unsigned* maxs = (unsigned*)(sd + PIX);
    float* acc     = (float*)(maxs + Nn);

    k_init   <<<1, 64, 0, stream>>>(maxs, acc);
    k_row_edt<<<Nn * Hh, Ww, 0, stream>>>(seg_gt, g2_in, g2_out);
    k_minplus<<<Nn * 16 * 64, 256, 0, stream>>>(g2_in, g2_out, sd, maxs);
    k_loss   <<<Nn * 256, 256, 0, stream>>>(seg_pred, seg_gt, edge_pred, edge_gt,
                                            sd, maxs, acc);
    k_final  <<<1, 64, 0, stream>>>(acc, out);
}